// GraphConvolution_55430847922769
// MI455X (gfx1250) — compile-verified
//
#include <hip/hip_runtime.h>

#define N_NODES_C 100000
#define IN_F 256
#define OUT_F 128

typedef __attribute__((ext_vector_type(2))) float v2f;
typedef __attribute__((ext_vector_type(8))) float v8f;

#define TILE_ROWS 64
#define TILE_K 32
#define NSTAGES (IN_F / TILE_K)   // 8
#define LDX (TILE_K + 4)          // 36 dwords: 36*row mod 64 is a permutation -> conflict-free A b64 reads
#define PSTR (OUT_F + 16)         // 144 float2 = 288 dwords, %64 = 32 -> lane halves hit disjoint banks

// -------------------------------------------------------------------------
// Kernel 1: support = X @ W  (fp32 WMMA 16x16x4), double-buffered LDS with
// async global->LDS copies for the X tile (ASYNCcnt) and register-staged W.
// Block = 256 threads = 8 waves; block tile 64 rows x 128 cols.
// -------------------------------------------------------------------------
__global__ __launch_bounds__(256)
void gcn_gemm_xw(const float* __restrict__ x, const float* __restrict__ w,
                 float* __restrict__ support, int n_nodes) {
  __shared__ float Xs[2][TILE_ROWS * LDX];        // 2 x 9.2 KB
  __shared__ v2f   Wp[2][(TILE_K / 2) * PSTR];    // 2 x 18.4 KB

  const int tid  = threadIdx.x;
  const int lane = tid & 31;
  const int wv   = tid >> 5;   // 0..7
  const int wr   = wv >> 1;    // row sub-tile 0..3
  const int wc   = wv & 1;     // col half 0..1
  const int block_row = blockIdx.x * TILE_ROWS;

  // ---- async X tile stage: 2 x b128 per thread (64 rows x 32 k) ----
  auto issue_x_async = [&](int k0, int buf) {
#pragma unroll
    for (int j = 0; j < 2; ++j) {
      int i  = tid + j * 256;      // 0..511 quad index
      int tr = i >> 3;             // row 0..63
      int tk = (i & 7) * 4;        // k quad 0..28
      int gr = block_row + tr;
      if (gr >= n_nodes) gr = n_nodes - 1;   // clamp tail (stores guarded)
      const float* gsrc = x + (size_t)gr * IN_F + (k0 + tk);
      unsigned ldsa = (unsigned)(size_t)&Xs[buf][tr * LDX + tk];
      asm volatile("global_load_async_to_lds_b128 %0, %1, off"
                   :: "v"(ldsa), "v"(gsrc)
                   : "memory");
    }
  };

  // ---- W tile stage via registers: 8 v2f per thread (32 k x 128 cols) ----
  auto load_w_regs = [&](int k0, v2f* r) {
#pragma unroll
    for (int j = 0; j < 8; ++j) {
      int i = tid + j * 256;       // 0..2047
      int p = i >> 7;              // k-pair 0..15
      int c = i & 127;
      size_t g = (size_t)(k0 + 2 * p) * OUT_F + c;
      r[j].x = w[g];
      r[j].y = w[g + OUT_F];
    }
  };
  auto store_w_lds = [&](int buf, const v2f* r) {
#pragma unroll
    for (int j = 0; j < 8; ++j) {
      int i = tid + j * 256;
      int p = i >> 7;
      int c = i & 127;
      Wp[buf][p * PSTR + c] = r[j];
    }
  };

  v8f acc0 = {}, acc1 = {}, acc2 = {}, acc3 = {};
  v2f wreg[8];

  // Fragment addressing per ISA layout:
  //  A 16x4: lanes 0-15 -> K=k+0/k+1, lanes 16-31 -> K=k+2/k+3
  //  B 4x16: lane%16 = N column, pair index p = k/2 + lane/16
  const int arow = wr * 16 + (lane & 15);
  const int kh   = (lane >> 4);
  const int bcol = wc * 64 + (lane & 15);

  // ---- prologue: stage 0 ----
  issue_x_async(0, 0);
  load_w_regs(0, wreg);
  store_w_lds(0, wreg);
  asm volatile("s_wait_asynccnt 0x0" ::: "memory");
  __syncthreads();

  for (int s = 0; s < NSTAGES; ++s) {
    const int cur = s & 1;
    const int nxt = cur ^ 1;

    if (s + 1 < NSTAGES) {
      issue_x_async((s + 1) * TILE_K, nxt);   // async, overlaps compute below
      load_w_regs((s + 1) * TILE_K, wreg);    // global loads in flight during compute
    }

#pragma unroll 4
    for (int k = 0; k < TILE_K; k += 4) {
      v2f a = *(const v2f*)&Xs[cur][arow * LDX + k + 2 * kh];

      const v2f* wp = &Wp[cur][((k >> 1) + kh) * PSTR + bcol];
      v2f b0 = wp[0];
      v2f b1 = wp[16];
      v2f b2 = wp[32];
      v2f b3 = wp[48];

      acc0 = __builtin_amdgcn_wmma_f32_16x16x4_f32(false, a, false, b0, (short)0, acc0, false, false);
      acc1 = __builtin_amdgcn_wmma_f32_16x16x4_f32(false, a, false, b1, (short)0, acc1, false, false);
      acc2 = __builtin_amdgcn_wmma_f32_16x16x4_f32(false, a, false, b2, (short)0, acc2, false, false);
      acc3 = __builtin_amdgcn_wmma_f32_16x16x4_f32(false, a, false, b3, (short)0, acc3, false, false);
    }
    __syncthreads();   // all waves done reading buffers of stage s

    if (s + 1 < NSTAGES) {
      store_w_lds(nxt, wreg);                            // W regs -> LDS
      asm volatile("s_wait_asynccnt 0x0" ::: "memory");  // X tile landed
      __syncthreads();                                   // publish stage s+1
    }
  }

  // Store D: VGPR r holds M=r (lanes 0-15) / M=8+r (lanes 16-31), N = lane%16
  const int col0  = wc * 64 + (lane & 15);
  const int rbase = block_row + wr * 16 + (lane >> 4) * 8;
#pragma unroll
  for (int r = 0; r < 8; ++r) {
    int grow = rbase + r;
    if (grow < n_nodes) {
      size_t o = (size_t)grow * OUT_F + col0;
      support[o +  0] = acc0[r];
      support[o + 16] = acc1[r];
      support[o + 32] = acc2[r];
      support[o + 48] = acc3[r];
    }
  }
}

// -------------------------------------------------------------------------
// Kernel 2: CSR row_ptr from sorted edge_row (segment boundaries)
// -------------------------------------------------------------------------
__global__ void gcn_build_row_ptr(const int* __restrict__ edge_row,
                                  int* __restrict__ row_ptr,
                                  int n_edges, int n_nodes) {
  int e = blockIdx.x * blockDim.x + threadIdx.x;
  if (e >= n_edges) return;
  int r = edge_row[e];
  int prev = (e == 0) ? -1 : edge_row[e - 1];
  for (int row = prev + 1; row <= r; ++row) row_ptr[row] = e;
  if (e == n_edges - 1) {
    for (int row = r + 1; row <= n_nodes; ++row) row_ptr[row] = n_edges;
  }
}

// -------------------------------------------------------------------------
// Kernel 3: out[row] = sum_{e in row} val[e] * support[col[e]] + bias
// One wave32 per node; lane owns float4 -> each edge gather is one coalesced
// 512B row read (support is L2-resident: 51MB < 192MB L2). 2-edge unroll.
// -------------------------------------------------------------------------
__global__ __launch_bounds__(256)
void gcn_spmm(const float* __restrict__ support, const int* __restrict__ row_ptr,
              const int* __restrict__ edge_col, const float* __restrict__ edge_val,
              const float* __restrict__ bias, float* __restrict__ out, int n_nodes) {
  int wave = (blockIdx.x * blockDim.x + threadIdx.x) >> 5;
  int lane = threadIdx.x & 31;
  if (wave >= n_nodes) return;

  int start = row_ptr[wave];
  int end   = row_ptr[wave + 1];

  float4 acc = make_float4(0.f, 0.f, 0.f, 0.f);
  int e = start;
  for (; e + 1 < end; e += 2) {
    int   c0 = edge_col[e];
    int   c1 = edge_col[e + 1];
    float v0 = edge_val[e];
    float v1 = edge_val[e + 1];
    float4 s0 = ((const float4*)(support + (size_t)c0 * OUT_F))[lane];
    float4 s1 = ((const float4*)(support + (size_t)c1 * OUT_F))[lane];
    acc.x += v0 * s0.x + v1 * s1.x;
    acc.y += v0 * s0.y + v1 * s1.y;
    acc.z += v0 * s0.z + v1 * s1.z;
    acc.w += v0 * s0.w + v1 * s1.w;
  }
  if (e < end) {
    int   c = edge_col[e];
    float v = edge_val[e];
    float4 s = ((const float4*)(support + (size_t)c * OUT_F))[lane];
    acc.x += v * s.x; acc.y += v * s.y; acc.z += v * s.z; acc.w += v * s.w;
  }

  float4 b = ((const float4*)bias)[lane];
  float4 r = make_float4(acc.x + b.x, acc.y + b.y, acc.z + b.z, acc.w + b.w);
  ((float4*)(out + (size_t)wave * OUT_F))[lane] = r;
}

// -------------------------------------------------------------------------
extern "C" void kernel_launch(void* const* d_in, const int* in_sizes, int n_in,
                              void* d_out, int out_size, void* d_ws, size_t ws_size,
                              hipStream_t stream) {
  const float* x        = (const float*)d_in[0];
  const int*   edge_row = (const int*)  d_in[1];
  const int*   edge_col = (const int*)  d_in[2];
  const float* edge_val = (const float*)d_in[3];
  const float* weight   = (const float*)d_in[4];
  const float* bias     = (const float*)d_in[5];
  float*       out      = (float*)d_out;

  const int n_nodes = N_NODES_C;
  const int n_edges = in_sizes[1];

  // Workspace layout: support [N*128 f32] then row_ptr [N+1 i32]
  float* support = (float*)d_ws;
  int*   row_ptr = (int*)((char*)d_ws + (size_t)n_nodes * OUT_F * sizeof(float));

  gcn_gemm_xw<<<(n_nodes + TILE_ROWS - 1) / TILE_ROWS, 256, 0, stream>>>(
      x, weight, support, n_nodes);

  gcn_build_row_ptr<<<(n_edges + 255) / 256, 256, 0, stream>>>(
      edge_row, row_ptr, n_edges, n_nodes);

  // one wave per node: n_nodes*32 threads
  long long threads_total = (long long)n_nodes * 32;
  int blocks = (int)((threads_total + 255) / 256);
  gcn_spmm<<<blocks, 256, 0, stream>>>(
      support, row_ptr, edge_col, edge_val, bias, out, n_nodes);
}